// Renderer_31610959298865
// MI455X (gfx1250) — compile-verified
//
#include <hip/hip_runtime.h>
#include <math.h>

typedef __attribute__((ext_vector_type(2))) float v2f;
typedef __attribute__((ext_vector_type(8))) float v8f;

#define SB     4
#define NSAMP  64
#define RHW    64

// One workgroup = 128 threads = 4 waves; each wave owns 16 rays.
// 16384 rays total -> 256 workgroups.
__global__ __launch_bounds__(128) void nerf_fused_kernel(
    const float* __restrict__ images,      // [4][3][3][64][64]
    const float* __restrict__ target_rays, // [4][1][64][64][8]
    const float* __restrict__ W_enc,       // [3][1024]
    const float* __restrict__ b_enc,       // [1024]
    const float* __restrict__ W1,          // [16][64]
    const float* __restrict__ b1,          // [64]
    const float* __restrict__ W2,          // [64][64]
    const float* __restrict__ b2,          // [64]
    const float* __restrict__ W3,          // [64][4]
    const float* __restrict__ b3,          // [4]
    float* __restrict__ out)               // rgb [4][64][64][3] ++ depth [4][64][64]
{
  __shared__ float sW1T[64][17];       // W1^T  [n][k]
  __shared__ float sW2T[64][65];       // W2^T  [n][k]
  __shared__ float sW3T[16][65];       // W3^T  [n][k], rows n>=4 zero-padded
  __shared__ float sB1[64];
  __shared__ float sB2[64];
  __shared__ float sB3[16];
  __shared__ float sAct[4][16][65];    // per-wave 16x64 activation tile
  __shared__ float sTile[4][16][17];   // per-wave feats (16x16) / raw (16x4) tile

  const int tid = threadIdx.x;
  // ---- stage weights (transposed) into LDS, once per workgroup ----
  for (int i = tid; i < 64 * 16; i += 128) { int n = i >> 4, k = i & 15; sW1T[n][k] = W1[k * 64 + n]; }
  for (int i = tid; i < 64 * 64; i += 128) { int n = i >> 6, k = i & 63; sW2T[n][k] = W2[k * 64 + n]; }
  for (int i = tid; i < 16 * 64; i += 128) { int n = i >> 6, k = i & 63; sW3T[n][k] = (n < 4) ? W3[k * 4 + n] : 0.0f; }
  if (tid < 64) { sB1[tid] = b1[tid]; sB2[tid] = b2[tid]; }
  if (tid < 16) sB3[tid] = (tid < 4) ? b3[tid] : 0.0f;
  __syncthreads();

  const int wv   = tid >> 5;
  const int lane = tid & 31;
  const int half = lane >> 4;   // which 16-lane half of the wave32
  const int m    = lane & 15;   // row (ray) / column index within fragments

  const int R  = blockIdx.x * 64 + wv * 16 + m;  // global ray id: (s, ry, rx)
  const int s  = R >> 12;
  const int ry = (R >> 6) & 63;
  const int rx = R & 63;

  // rays used for SAMPLING: batch s  (target_rays broadcast over MB)
  const float* tr = target_rays + (size_t)((s * RHW + ry) * RHW + rx) * 8;
  const float ox = tr[0], oy = tr[1], oz = tr[2];
  const float dx = tr[3], dy = tr[4], dz = tr[5];
  const float nr = tr[6], fr = tr[7];

  // rays used for COMPOSITING z/deltas: reference uses z[:sb] => flattened
  // batch index s => target_rays[s//2]
  const int s2 = s >> 1;
  const float* tr2 = target_rays + (size_t)((s2 * RHW + ry) * RHW + rx) * 8;
  const float nrC  = tr2[6], frC = tr2[7];
  const float dltC = (frC - nrC) * (1.0f / NSAMP);

  float T = 1.0f;
  float accR = 0.0f, accG = 0.0f, accB = 0.0f, accD = 0.0f;

  const float* imgS = images + (size_t)s * 3 * 3 * 4096;  // images[s][mb][k][y][x], mb<2 used

  for (int ns = 0; ns < NSAMP; ++ns) {
    const float t = (ns + 0.5f) * (1.0f / NSAMP);

    // ================= on-the-fly trilinear feature sampling =================
    // vol[c, z, y, x] = img(y,x) . W_enc[:, c*64+z] + b_enc[c*64+z]
    // => bilinear-blend the pixel first, then dot with the two z-columns.
    {
      const float zs = nr + (fr - nr) * t;
      float gx = (ox + dx * zs + 1.0f) * 0.5f * 63.0f;
      float gy = (oy + dy * zs + 1.0f) * 0.5f * 63.0f;
      float gz = (oz + dz * zs + 1.0f) * 0.5f * 63.0f;
      gx = fminf(fmaxf(gx, 0.0f), 63.0f);
      gy = fminf(fmaxf(gy, 0.0f), 63.0f);
      gz = fminf(fmaxf(gz, 0.0f), 63.0f);
      const int x0 = (int)gx, y0 = (int)gy, z0 = (int)gz;
      const int x1 = (x0 + 1 > 63) ? 63 : x0 + 1;
      const int y1 = (y0 + 1 > 63) ? 63 : y0 + 1;
      const int z1 = (z0 + 1 > 63) ? 63 : z0 + 1;
      const float fx = gx - (float)x0, fy = gy - (float)y0, fz = gz - (float)z0;
      const float w00 = (1.0f - fx) * (1.0f - fy), w01 = fx * (1.0f - fy);
      const float w10 = (1.0f - fx) * fy,          w11 = fx * fy;
      // pixel blend, summed over the MB=2 views (mean folded in later as 0.5)
      float ps[3];
      for (int k = 0; k < 3; ++k) {
        float sum = 0.0f;
        for (int mb = 0; mb < 2; ++mb) {
          const float* ib = imgS + (size_t)(mb * 3 + k) * 4096;
          sum += w00 * ib[y0 * 64 + x0] + w01 * ib[y0 * 64 + x1] +
                 w10 * ib[y1 * 64 + x0] + w11 * ib[y1 * 64 + x1];
        }
        ps[k] = sum;
      }
      // this lane produces 8 of the 16 channels for its ray
      for (int c8 = 0; c8 < 8; ++c8) {
        const int c = half * 8 + c8;
        const int col0 = c * 64 + z0, col1 = c * 64 + z1;
        const float d0 = 0.5f * (ps[0] * W_enc[col0] + ps[1] * W_enc[1024 + col0] +
                                 ps[2] * W_enc[2048 + col0]) + b_enc[col0];
        const float d1 = 0.5f * (ps[0] * W_enc[col1] + ps[1] * W_enc[1024 + col1] +
                                 ps[2] * W_enc[2048 + col1]) + b_enc[col1];
        sTile[wv][m][c] = (1.0f - fz) * d0 + fz * d1;
      }
    }

    // ================= layer 1: [16x16] @ [16x64] via 16 WMMA ================
    v8f h[4];
    for (int n = 0; n < 4; ++n) { const float bb = sB1[n * 16 + m]; for (int r = 0; r < 8; ++r) h[n][r] = bb; }
    for (int k = 0; k < 4; ++k) {
      v2f a; a[0] = sTile[wv][m][4 * k + 2 * half]; a[1] = sTile[wv][m][4 * k + 2 * half + 1];
      for (int n = 0; n < 4; ++n) {
        v2f b; b[0] = sW1T[n * 16 + m][4 * k + 2 * half]; b[1] = sW1T[n * 16 + m][4 * k + 2 * half + 1];
        h[n] = __builtin_amdgcn_wmma_f32_16x16x4_f32(false, a, false, b, (short)0, h[n], false, false);
      }
    }
    for (int n = 0; n < 4; ++n)
      for (int r = 0; r < 8; ++r)
        sAct[wv][r + 8 * half][n * 16 + m] = fmaxf(h[n][r], 0.0f);

    // ================= layer 2: [16x64] @ [64x64] via 64 WMMA ================
    v8f h2[4];
    for (int n = 0; n < 4; ++n) { const float bb = sB2[n * 16 + m]; for (int r = 0; r < 8; ++r) h2[n][r] = bb; }
    for (int k = 0; k < 16; ++k) {
      v2f a; a[0] = sAct[wv][m][4 * k + 2 * half]; a[1] = sAct[wv][m][4 * k + 2 * half + 1];
      for (int n = 0; n < 4; ++n) {
        v2f b; b[0] = sW2T[n * 16 + m][4 * k + 2 * half]; b[1] = sW2T[n * 16 + m][4 * k + 2 * half + 1];
        h2[n] = __builtin_amdgcn_wmma_f32_16x16x4_f32(false, a, false, b, (short)0, h2[n], false, false);
      }
    }
    for (int n = 0; n < 4; ++n)
      for (int r = 0; r < 8; ++r)
        sAct[wv][r + 8 * half][n * 16 + m] = fmaxf(h2[n][r], 0.0f);

    // ============ layer 3: [16x64] @ [64x16] (4 valid cols) 16 WMMA ==========
    v8f h3;
    { const float bb = sB3[m]; for (int r = 0; r < 8; ++r) h3[r] = bb; }
    for (int k = 0; k < 16; ++k) {
      v2f a; a[0] = sAct[wv][m][4 * k + 2 * half]; a[1] = sAct[wv][m][4 * k + 2 * half + 1];
      v2f b; b[0] = sW3T[m][4 * k + 2 * half];     b[1] = sW3T[m][4 * k + 2 * half + 1];
      h3 = __builtin_amdgcn_wmma_f32_16x16x4_f32(false, a, false, b, (short)0, h3, false, false);
    }
    if (m < 4)  // scatter raw[row][m] for the 4 valid output columns
      for (int r = 0; r < 8; ++r)
        sTile[wv][r + 8 * half][m] = h3[r];

    // ======================= volume-render compositing =======================
    {
      const float r0 = sTile[wv][m][0];
      const float r1 = sTile[wv][m][1];
      const float r2 = sTile[wv][m][2];
      const float r3 = sTile[wv][m][3];
      const float sg    = fmaxf(r3, 0.0f);
      const float delta = (ns == NSAMP - 1) ? 10000.0f : dltC;
      const float alpha = 1.0f - expf(-delta * sg);
      const float w     = alpha * T;
      T = T * (1.0f - alpha + 1e-10f);
      accR += w / (1.0f + expf(-r0));
      accG += w / (1.0f + expf(-r1));
      accB += w / (1.0f + expf(-r2));
      accD += w * (nrC + (frC - nrC) * t);
    }
  }

  if (half == 0) {
    const int pix = (s * RHW + ry) * RHW + rx;
    out[pix * 3 + 0] = accR;
    out[pix * 3 + 1] = accG;
    out[pix * 3 + 2] = accB;
    out[SB * RHW * RHW * 3 + pix] = accD;
  }
}

extern "C" void kernel_launch(void* const* d_in, const int* in_sizes, int n_in,
                              void* d_out, int out_size, void* d_ws, size_t ws_size,
                              hipStream_t stream) {
  (void)in_sizes; (void)n_in; (void)out_size; (void)d_ws; (void)ws_size;
  const float* images      = (const float*)d_in[0];
  const float* target_rays = (const float*)d_in[1];
  const float* W_enc       = (const float*)d_in[2];
  const float* b_enc       = (const float*)d_in[3];
  const float* W1          = (const float*)d_in[4];
  const float* b1          = (const float*)d_in[5];
  const float* W2          = (const float*)d_in[6];
  const float* b2          = (const float*)d_in[7];
  const float* W3          = (const float*)d_in[8];
  const float* b3          = (const float*)d_in[9];
  float* out = (float*)d_out;

  // 16384 rays / 64 rays per workgroup
  nerf_fused_kernel<<<256, 128, 0, stream>>>(
      images, target_rays, W_enc, b_enc, W1, b1, W2, b2, W3, b3, out);
}